// Chamfer_481036337229
// MI455X (gfx1250) — compile-verified
//
#include <hip/hip_runtime.h>

typedef __attribute__((ext_vector_type(2))) float v2f;
typedef __attribute__((ext_vector_type(8))) float v8f;

#define WG_THREADS 256
#define ROWS_PER_WG 128   // 8 waves * 16 rows per wave
#define MAX_TGT 4096

// One direction of chamfer: for each point in P (rows), min squared distance
// over all points in Q (cols). d = |p|^2 + |q|^2 - 2 p.q.
// WMMA f32 16x16x4 with augmented K=4:
//   A row  = [px, py, pz, 1]
//   B col  = [-2qx, -2qy, -2qz, |q|^2]
//   result = |q|^2 - 2 p.q   (add |p|^2 per row after the column-min)
__global__ __launch_bounds__(WG_THREADS)
void chamfer_dir_kernel(const float* __restrict__ P, const float* __restrict__ Q,
                        float* __restrict__ minOut, int Np, int Nq)
{
    __shared__ float4 sQ[MAX_TGT];   // 64 KB: (-2qx,-2qy,-2qz,|q|^2) per target

    const int b    = blockIdx.y;           // batch
    const int tile = blockIdx.x;           // 128-row tile within batch
    const int t    = threadIdx.x;
    const int lane = t & 31;
    const int wave = t >> 5;

    const float* Pb = P + (size_t)b * Np * 3;
    const float* Qb = Q + (size_t)b * Nq * 3;

    // Stage transformed target set into LDS (all 8 waves share it).
    for (int n = t; n < Nq; n += WG_THREADS) {
        float qx = Qb[3*n+0], qy = Qb[3*n+1], qz = Qb[3*n+2];
        sQ[n] = make_float4(-2.0f*qx, -2.0f*qy, -2.0f*qz, qx*qx + qy*qy + qz*qz);
    }
    __syncthreads();

    // Build A fragment: 16 query rows per wave.
    // A layout (32-bit 16x4): VGPR0 = K0 (lanes 0-15) / K2 (lanes 16-31),
    //                         VGPR1 = K1 (lanes 0-15) / K3 (lanes 16-31).
    const int m0 = tile * ROWS_PER_WG + wave * 16;
    const int m  = m0 + (lane & 15);
    float px = Pb[3*m+0], py = Pb[3*m+1], pz = Pb[3*m+2];
    const float p2 = px*px + py*py + pz*pz;
    const bool lo16 = lane < 16;
    v2f afrag;
    afrag.x = lo16 ? px : pz;     // K0 : K2
    afrag.y = lo16 ? py : 1.0f;   // K1 : K3

    // B fragment source: lanes 0-15 read (-2qx,-2qy) = rows K0,K1 of column n;
    // lanes 16-31 read (-2qz,|q|^2) = rows K2,K3. One ds_load_b64 per chunk.
    const float2* qptr = reinterpret_cast<const float2*>(
        reinterpret_cast<const float*>(sQ) + ((lane & 15) << 2) + ((lane >> 4) << 1));

    float rowmin[8];
#pragma unroll
    for (int r = 0; r < 8; ++r) rowmin[r] = 3.4e38f;

    const v8f czero = {};
    const int chunks = Nq >> 4;
#pragma unroll 4
    for (int c = 0; c < chunks; ++c) {
        float2 bq = qptr[(size_t)c << 5];   // advance 16 targets = 256B per chunk
        v2f bfrag;
        bfrag.x = bq.x;
        bfrag.y = bq.y;
        v8f d = __builtin_amdgcn_wmma_f32_16x16x4_f32(
            false, afrag, false, bfrag, (short)0, czero, false, false);
#pragma unroll
        for (int r = 0; r < 8; ++r) rowmin[r] = fminf(rowmin[r], d[r]);
    }

    // C/D layout: VGPR r holds row r (lanes 0-15) and row r+8 (lanes 16-31),
    // N = lane & 15. Min across columns = min across each 16-lane half.
    float* outB = minOut + (size_t)b * Np + m0;
#pragma unroll
    for (int r = 0; r < 8; ++r) {
        float v = rowmin[r];
        v = fminf(v, __shfl_xor(v, 1, 32));
        v = fminf(v, __shfl_xor(v, 2, 32));
        v = fminf(v, __shfl_xor(v, 4, 32));
        v = fminf(v, __shfl_xor(v, 8, 32));
        float vlo = __shfl(v, 0, 32);    // row r min
        float vhi = __shfl(v, 16, 32);   // row r+8 min
        // lane r holds p2 of row r; lane r+8 holds p2 of row r+8
        if (lane == r)          outB[r]     = vlo + p2;
        else if (lane == r + 8) outB[r + 8] = vhi + p2;
    }
}

// Deterministic single-block reduction: out = mean(minL) + mean(minR)
__global__ __launch_bounds__(WG_THREADS)
void chamfer_reduce_kernel(const float* __restrict__ mins, int nL, int nR,
                           float* __restrict__ out)
{
    __shared__ float red[WG_THREADS];
    const int t = threadIdx.x;
    float accL = 0.0f, accR = 0.0f;
    for (int i = t; i < nL; i += WG_THREADS) accL += mins[i];
    for (int i = t; i < nR; i += WG_THREADS) accR += mins[nL + i];
    red[t] = accL / (float)nL + accR / (float)nR;
    __syncthreads();
    for (int s = WG_THREADS / 2; s > 0; s >>= 1) {
        if (t < s) red[t] += red[t + s];
        __syncthreads();
    }
    if (t == 0) out[0] = red[0];
}

extern "C" void kernel_launch(void* const* d_in, const int* in_sizes, int n_in,
                              void* d_out, int out_size, void* d_ws, size_t ws_size,
                              hipStream_t stream)
{
    const float* x = (const float*)d_in[0];   // [B, N, 3] f32
    const float* y = (const float*)d_in[1];   // [B, M, 3] f32
    float* out = (float*)d_out;               // scalar f32

    const int B = 16;
    const int N = in_sizes[0] / (3 * B);      // 4096
    const int M = in_sizes[1] / (3 * B);      // 4096

    float* minL = (float*)d_ws;               // [B*N]
    float* minR = minL + (size_t)B * N;       // [B*M]

    dim3 blk(WG_THREADS);
    dim3 gridL(N / ROWS_PER_WG, B);
    dim3 gridR(M / ROWS_PER_WG, B);

    hipLaunchKernelGGL(chamfer_dir_kernel, gridL, blk, 0, stream, x, y, minL, N, M);
    hipLaunchKernelGGL(chamfer_dir_kernel, gridR, blk, 0, stream, y, x, minR, M, N);
    hipLaunchKernelGGL(chamfer_reduce_kernel, dim3(1), blk, 0, stream,
                       minL, B * N, B * M, out);
}